// Aggregator_1846835937457
// MI455X (gfx1250) — compile-verified
//
#include <hip/hip_runtime.h>

// Mean over neighbor axis: mailbox[50000, 32, 128] f32 -> out[50000, 128] f32.
// Memory-bound (~845 MB @ 23.3 TB/s ~= 36 us). Strategy:
//  - 1 wave per node, 8 waves/block, 6250 blocks (50000/8 exact -> full waves,
//    EXEC all-ones for WMMA).
//  - Accumulate with V_WMMA_F32_16X16X4_F32: A = broadcast(1/32), B = 4 neighbor
//    rows x 16 features, C chained -> D row 0 holds the running mean. Matrix pipe
//    does the reduction; VALU stays free; loads are the only bottleneck.
//  - Non-temporal loads/stores: 819 MB streamed once >> 192 MB L2.

typedef __attribute__((ext_vector_type(2))) float v2f;
typedef __attribute__((ext_vector_type(8))) float v8f;

#define N_NODES 50000
#define MAX_DEG 32
#define FEAT    128

__global__ __launch_bounds__(256) void mean_aggr_wmma(
    const float* __restrict__ mailbox, float* __restrict__ out) {
  const int lane = threadIdx.x & 31;
  const int wave = threadIdx.x >> 5;
  const int node = blockIdx.x * 8 + wave;   // grid sized so node < N_NODES always

  const float* nodeBase = mailbox + (size_t)node * (MAX_DEG * FEAT);
  const int half = lane >> 4;     // 0: lanes 0-15, 1: lanes 16-31
  const int col  = lane & 15;     // N index within 16-feature chunk

  // A (16x4) = 1/32 everywhere: every entry identical, so each lane's 2 A-VGPRs
  // are just the constant. D = A*B + C computes the running mean over K.
  v2f a;
  a[0] = 1.0f / 32.0f;
  a[1] = 1.0f / 32.0f;

  // Process 8 chunks of 16 features; within a chunk, chain 8 WMMAs over K=32.
  for (int fc = 0; fc < FEAT / 16; ++fc) {
    v8f acc = {};  // C starts at zero

    // B layout (4x16 f32, 2 VGPRs):
    //   V0: lanes 0-15 = row K=0, lanes 16-31 = row K=2
    //   V1: lanes 0-15 = row K=1, lanes 16-31 = row K=3
    const float* p = nodeBase + (size_t)(half * 2) * FEAT + fc * 16 + col;

#pragma unroll
    for (int kc = 0; kc < MAX_DEG / 4; ++kc) {
      v2f b;
      b[0] = __builtin_nontemporal_load(p);         // row kc*4 + 2*half
      b[1] = __builtin_nontemporal_load(p + FEAT);  // row kc*4 + 2*half + 1
      acc = __builtin_amdgcn_wmma_f32_16x16x4_f32(
          /*neg_a=*/false, a, /*neg_b=*/false, b,
          /*c_mod=*/(short)0, acc, /*reuse_a=*/false, /*reuse_b=*/false);
      p += 4 * FEAT;
    }

    // All rows of D are identical (broadcast A). Row M=0 lives in acc[0],
    // lanes 0-15 holding N=0..15. Half-wave 64B store; stores are 3% of traffic.
    if (lane < 16) {
      __builtin_nontemporal_store(acc[0],
          out + (size_t)node * FEAT + fc * 16 + lane);
    }
  }
}

extern "C" void kernel_launch(void* const* d_in, const int* in_sizes, int n_in,
                              void* d_out, int out_size, void* d_ws, size_t ws_size,
                              hipStream_t stream) {
  const float* mailbox = (const float*)d_in[0];
  float* out = (float*)d_out;
  (void)in_sizes; (void)n_in; (void)out_size; (void)d_ws; (void)ws_size;

  dim3 grid(N_NODES / 8);  // 6250 blocks, 8 full waves each
  dim3 block(256);
  mean_aggr_wmma<<<grid, block, 0, stream>>>(mailbox, out);
}